// MMI_softmax_Loss_70566312673356
// MI455X (gfx1250) — compile-verified
//
#include <hip/hip_runtime.h>
#include <hip/hip_bf16.h>

// Reference shapes: embeddings (10240, 32000) f32, targets (2048,) i32.
// out = -mean_i( emb[i, t_i] / sum_{k=1..4} emb[i + 2048k, t_i] )
// Pure gather problem: 5 scalar loads per target row, then a global reduction.

#define N_CLASSES 32000

typedef __attribute__((ext_vector_type(2))) float v2f;
typedef __attribute__((ext_vector_type(8))) float v8f;

// Stage 1: each thread handles one target row; per-block partial sum -> d_ws.
__global__ void __launch_bounds__(256)
mmi_pick_partial(const float* __restrict__ emb,
                 const int* __restrict__ targets,
                 float* __restrict__ block_partials,
                 int dim, int n_groups) {
    const int i = blockIdx.x * blockDim.x + threadIdx.x;

    float p = 0.0f;
    if (i < dim) {
        const int t = targets[i];
        const size_t row_stride = (size_t)N_CLASSES;
        const float* base = emb + (size_t)i * row_stride + (size_t)t;
        const size_t group_off = (size_t)dim * row_stride;   // 2048*32000 elems

        // numerator: examples[i, t]
        const float num = base[0];

        // denominator: segment_sum of negatives at column t
        float den = 0.0f;
        #pragma unroll 4
        for (int k = 1; k <= n_groups; ++k) {
            den += base[(size_t)k * group_off];
        }
        p = num / den;
    }
    // control flow reconverged here: EXEC is all-1s for the WMMA below.

    // Intra-wave (32 lane) sum via V_WMMA_F32_16X16X4_F32.
    // A holds per-lane {p, 0}; B is all ones, so D[m,n] = rowsum_m(A) for
    // every n. Summing the 8 D VGPRs in lane L covers rows {0..7} (L<16) or
    // {8..15} (L>=16); adding the cross-half partner yields sum over all 16
    // rows = total of all A elements = sum of the 32 lane values. This is
    // independent of the exact A-slot->(m,k) mapping because B is uniform.
    v2f a; a.x = p;    a.y = 0.0f;
    v2f b; b.x = 1.0f; b.y = 1.0f;
    v8f acc = {};
    acc = __builtin_amdgcn_wmma_f32_16x16x4_f32(
        /*neg_a=*/false, a, /*neg_b=*/false, b,
        /*c_mod=*/(short)0, acc, /*reuse_a=*/false, /*reuse_b=*/false);

    float s = acc[0] + acc[1] + acc[2] + acc[3] +
              acc[4] + acc[5] + acc[6] + acc[7];
    s += __shfl_xor(s, 16, 32);          // combine the two half-waves

    __shared__ float wave_sums[8];       // 256 threads / wave32 = 8 waves
    const int wave = threadIdx.x >> 5;
    const int lane = threadIdx.x & 31;
    if (lane == 0) wave_sums[wave] = s;
    __syncthreads();

    if (threadIdx.x == 0) {
        float bs = 0.0f;
        const int nwaves = blockDim.x >> 5;
        for (int w = 0; w < nwaves; ++w) bs += wave_sums[w];
        block_partials[blockIdx.x] = bs;
    }
}

// Stage 2: deterministic fold of the block partials -> -sum/dim.
__global__ void mmi_finalize(const float* __restrict__ block_partials,
                             float* __restrict__ out,
                             int n_parts, float inv_dim) {
    if (blockIdx.x == 0 && threadIdx.x == 0) {
        float s = 0.0f;
        for (int i = 0; i < n_parts; ++i) s += block_partials[i];
        out[0] = -s * inv_dim;
    }
}

extern "C" void kernel_launch(void* const* d_in, const int* in_sizes, int n_in,
                              void* d_out, int out_size, void* d_ws, size_t ws_size,
                              hipStream_t stream) {
    const float* emb = (const float*)d_in[0];   // (10240, 32000) f32
    const int*   tgt = (const int*)d_in[1];     // (2048,) i32
    float* out = (float*)d_out;                 // scalar f32
    float* ws  = (float*)d_ws;                  // block partials

    const int dim      = in_sizes[1];                 // 2048
    const int n_rows   = in_sizes[0] / N_CLASSES;     // 10240
    const int n_groups = (n_rows - dim) / dim;        // 4

    const int threads = 256;                          // 8 waves (wave32)
    const int blocks  = (dim + threads - 1) / threads; // 8

    mmi_pick_partial<<<blocks, threads, 0, stream>>>(emb, tgt, ws, dim, n_groups);
    mmi_finalize<<<1, 32, 0, stream>>>(ws, out, blocks, 1.0f / (float)dim);
}